// AdjointODE_19464791786131
// MI455X (gfx1250) — compile-verified
//
#include <hip/hip_runtime.h>

typedef __attribute__((ext_vector_type(2))) float v2f;
typedef __attribute__((ext_vector_type(8))) float v8f;

#define BATCH   32768
#define DIM     128
#define HID     256
#define NSTEPS  50
#define WAVES   8
#define RPW     16              // rows per wave
#define RPWG    (WAVES * RPW)   // 128 rows per workgroup
#define SSTR    66              // stage row stride in floats (padded, even)
#define W1TS    (DIM + 2)       // transposed W1 row stride (130, even)
#define W2TS    (HID + 2)       // transposed W2 row stride (258, even)

#if __has_builtin(__builtin_amdgcn_tanhf)
  #define DEV_TANH(x) __builtin_amdgcn_tanhf(x)
#elif __has_builtin(__builtin_amdgcn_tanh_f32)
  #define DEV_TANH(x) __builtin_amdgcn_tanh_f32(x)
#else
  #define DEV_TANH(x) tanhf(x)
#endif

__global__ __launch_bounds__(256, 1)
void node_euler_f32wmma(const float* __restrict__ inp,   // [BATCH, DIM]
                        const float* __restrict__ ts,    // [NSTEPS+1]
                        const float* __restrict__ W1g,   // [DIM, HID] row-major
                        const float* __restrict__ b1g,   // [HID]
                        const float* __restrict__ W2g,   // [HID, DIM] row-major
                        const float* __restrict__ b2g,   // [DIM]
                        float* __restrict__ outp)        // [BATCH, DIM]
{
    // Weights stored TRANSPOSED in LDS: sW1t[n*W1TS + k] = W1[k][n]
    // so a B-fragment (W[k][n], W[k+1][n]) is one contiguous ds_load_b64.
    __shared__ __align__(16) float sW1t[HID * W1TS];   // 256 x 130
    __shared__ __align__(16) float sW2t[DIM * W2TS];   // 128 x 258
    __shared__ float sb1[HID];
    __shared__ float sb2[DIM];
    __shared__ __align__(16) float stage[WAVES * RPW * SSTR];

    const int tid  = threadIdx.x;
    const int wave = tid >> 5;
    const int lane = tid & 31;
    const int lm   = lane & 15;       // low lane index
    const int hi   = lane >> 4;       // half-wave select

    // Cooperative loads: coalesced global reads, transposed LDS writes (one-time)
    for (int i = tid; i < DIM * HID; i += 256) {
        const int k = i >> 8;         // i / HID
        const int n = i & (HID - 1);  // i % HID
        sW1t[n * W1TS + k] = W1g[i];
    }
    for (int i = tid; i < HID * DIM; i += 256) {
        const int k = i >> 7;         // i / DIM
        const int n = i & (DIM - 1);  // i % DIM
        sW2t[n * W2TS + k] = W2g[i];
    }
    if (tid < HID) sb1[tid] = b1g[tid];
    if (tid < DIM) sb2[tid] = b2g[tid];
    __syncthreads();

    float* st = &stage[wave * RPW * SSTR];      // wave-private stage region

    const int row = blockIdx.x * RPWG + wave * RPW + lm;

    // h resident in registers, A-fragment layout:
    // hA[j] holds (h[row][4j + 2*hi], h[row][4j + 2*hi + 1])
    v2f hA[DIM / 4];
#pragma unroll
    for (int j = 0; j < DIM / 4; ++j)
        hA[j] = *(const v2f*)&inp[row * DIM + 4 * j + 2 * hi];

#pragma unroll 1
    for (int s = 0; s < NSTEPS; ++s) {
        __syncthreads();                         // keeps LDS loads inside the step loop
        const float dt = ts[s + 1] - ts[s];

        // acc2: 8 C/D tiles (16x16) covering out columns 0..127, init with b2
        v8f acc2[DIM / 16];
#pragma unroll
        for (int t = 0; t < DIM / 16; ++t) {
            const float bv = sb2[t * 16 + lm];
#pragma unroll
            for (int v = 0; v < 8; ++v) acc2[t][v] = bv;
        }

#pragma unroll 1
        for (int c = 0; c < HID / 64; ++c) {     // 64-wide hidden-column chunks
            // ---- GEMM1 chunk: a_chunk = h @ W1[:, c*64 : c*64+64] + b1 ----
            v8f acc1[4];
#pragma unroll
            for (int t = 0; t < 4; ++t) {
                const float bv = sb1[c * 64 + t * 16 + lm];
#pragma unroll
                for (int v = 0; v < 8; ++v) acc1[t][v] = bv;
            }
#pragma unroll
            for (int j = 0; j < DIM / 4; ++j) {  // K = 128 in steps of 4
                const v2f av = hA[j];
                const int k0 = 4 * j + 2 * hi;
#pragma unroll
                for (int t = 0; t < 4; ++t) {
                    const int n = c * 64 + t * 16 + lm;
                    const v2f bv = *(const v2f*)&sW1t[n * W1TS + k0]; // ds_load_b64
                    acc1[t] = __builtin_amdgcn_wmma_f32_16x16x4_f32(
                        false, av, false, bv, (short)0, acc1[t], false, false);
                }
            }

            // ---- tanh, stage C/D-layout tiles as row-major a[m][k_local] ----
#pragma unroll
            for (int t = 0; t < 4; ++t)
#pragma unroll
                for (int v = 0; v < 8; ++v)
                    st[(v + 8 * hi) * SSTR + t * 16 + lm] = DEV_TANH(acc1[t][v]);
            // same-wave DS ops are in-order: reads below see the writes above

            // ---- GEMM2 partial: acc2 += a_chunk @ W2[c*64 : c*64+64, :] ----
#pragma unroll 2
            for (int j = 0; j < 16; ++j) {
                const v2f av = *(const v2f*)&st[lm * SSTR + 4 * j + 2 * hi];
                const int k0 = c * 64 + 4 * j + 2 * hi;
#pragma unroll
                for (int t = 0; t < DIM / 16; ++t) {
                    const int n = t * 16 + lm;
                    const v2f bv = *(const v2f*)&sW2t[n * W2TS + k0]; // ds_load_b64
                    acc2[t] = __builtin_amdgcn_wmma_f32_16x16x4_f32(
                        false, av, false, bv, (short)0, acc2[t], false, false);
                }
            }
        }

        // ---- h += dt * f(h): C/D layout -> A layout via per-tile LDS stage ----
#pragma unroll
        for (int t = 0; t < DIM / 16; ++t) {
#pragma unroll
            for (int v = 0; v < 8; ++v)
                st[(v + 8 * hi) * SSTR + lm] = dt * acc2[t][v];
#pragma unroll
            for (int j2 = 0; j2 < 4; ++j2) {
                const v2f d = *(const v2f*)&st[lm * SSTR + 4 * j2 + 2 * hi];
                hA[t * 4 + j2] += d;
            }
        }
    }

#pragma unroll
    for (int j = 0; j < DIM / 4; ++j)
        *(v2f*)&outp[row * DIM + 4 * j + 2 * hi] = hA[j];
}

extern "C" void kernel_launch(void* const* d_in, const int* in_sizes, int n_in,
                              void* d_out, int out_size, void* d_ws, size_t ws_size,
                              hipStream_t stream) {
    const float* inp = (const float*)d_in[0];   // inputs  [32768,128]
    const float* ts  = (const float*)d_in[1];   // timestamps [51]
    const float* W1  = (const float*)d_in[2];   // [128,256]
    const float* b1  = (const float*)d_in[3];   // [256]
    const float* W2  = (const float*)d_in[4];   // [256,128]
    const float* b2  = (const float*)d_in[5];   // [128]
    float* outp = (float*)d_out;

    node_euler_f32wmma<<<dim3(BATCH / RPWG), dim3(256), 0, stream>>>(
        inp, ts, W1, b1, W2, b2, outp);
}